// GFDS_32435593019878
// MI455X (gfx1250) — compile-verified
//
#include <hip/hip_runtime.h>

// SS2D (VMamba selective-scan 2D) for MI455X / gfx1250.
// Pipeline: [K1] x_proj GEMM via V_WMMA_F32_16X16X4_F32 (xs gathered on the fly,
//                weight tile staged zero-padded in LDS -> branch-free inner loop)
//           [K2] dt_proj GEMM via WMMA + bias + softplus -> delta
//           [K3/K4/K5] 3-phase chunked diagonal-linear scan (chunk transition
//                      factor = exp(A * sum(delta)) so phase 1 only needs the
//                      local zero-init state q and sum(delta))
//           phase 3 atomically merges (CrossMerge is an involutive permutation)
//           [K6] LayerNorm over channels.

typedef float v2f __attribute__((ext_vector_type(2)));
typedef float v8f __attribute__((ext_vector_type(8)));

#define BDIM   4
#define DDIM   96
#define LDIM   4096
#define NST    16
#define RRANK  6
#define CPAD   48      // 38 rows of x_dbl padded to 3 WMMA tiles
#define NCHUNK 32
#define TCHUNK 128

// xs[b,k,d,l] = x[b,d, xs_off(k,l)]   (H = W = 64, all maps are involutions,
// so the CrossMerge scatter target is the same xs_off(k,l))
__device__ __forceinline__ int xs_off(int k, int l) {
    switch (k & 3) {
        case 0:  return l;
        case 1:  return ((l & 63) << 6) | (l >> 6);
        case 2:  return (LDIM - 1) - l;
        default: { int lr = (LDIM - 1) - l; return ((lr & 63) << 6) | (lr >> 6); }
    }
}

__device__ __forceinline__ float softplus_f(float v) {
    return v > 20.f ? v : log1pf(__expf(v));
}

// ---------------- K1: x_dbl[b,k,c,l] = sum_d Wp[k,c,d] * xs[b,k,d,l] --------
// M = c (38 -> 3 tiles of 16), N = l (16-wide tiles), K-dim = d (24 steps of 4)
// A tile staged zero-padded in LDS -> unconditional loads in the WMMA loop.
__global__ __launch_bounds__(128) void ss2d_xproj_wmma(
    const float* __restrict__ x, const float* __restrict__ Wp,
    float* __restrict__ xdbl)
{
    __shared__ float Wlds[16][97];              // 97 stride: spread LDS banks
    const int tid  = threadIdx.x;
    const int lane = tid & 31;
    const int wave = tid >> 5;
    const int lt   = blockIdx.x * 4 + wave;     // l-tile 0..255
    const int c0   = blockIdx.y * 16;           // 0,16,32
    const int bk   = blockIdx.z;                // b*4+k
    const int b    = bk >> 2, k = bk & 3;
    const int l0   = lt * 16;
    const int m    = lane & 15;
    const int koff = (lane >> 4) * 2;           // f32 A/B: K = (lane/16)*2 + j
    const int l    = l0 + m;
    const int goff = xs_off(k, l);              // gather column from x

    // cooperative zero-padded weight tile load (rows c0..c0+15 of Wp[k])
    const float* Wk = Wp + (size_t)k * 38 * DDIM;
    for (int i = tid; i < 16 * DDIM; i += 128) {
        const int row = i / DDIM, col = i - row * DDIM;
        const int c = c0 + row;
        Wlds[row][col] = (c < 38) ? Wk[c * DDIM + col] : 0.f;
    }
    __syncthreads();

    const float* xb = x + (size_t)b * DDIM * LDIM;
    v8f acc = {};
    #pragma unroll 4
    for (int kk = 0; kk < 24; ++kk) {
        const int d = kk * 4 + koff;
        v2f av, bv;
        av.x = Wlds[m][d];
        av.y = Wlds[m][d + 1];
        bv.x = xb[(size_t)d       * LDIM + goff];
        bv.y = xb[(size_t)(d + 1) * LDIM + goff];
        acc = __builtin_amdgcn_wmma_f32_16x16x4_f32(false, av, false, bv,
                                                    (short)0, acc, false, false);
    }
    float* orow = xdbl + (size_t)bk * CPAD * LDIM;
    const int mhi = (lane >> 4) << 3;           // C: M = r + 8*(lane/16)
    #pragma unroll
    for (int r = 0; r < 8; ++r)
        orow[(size_t)(c0 + r + mhi) * LDIM + l0 + m] = acc[r];
}

// ---------------- K2: delta = softplus( Wdt[k] @ x_dbl[rows 0..5] + bias ) ---
// A tile (16 x R=6, padded to 8) staged zero-padded in LDS. B loads rows 0..7
// of padded x_dbl unconditionally: rows 6,7 multiply zeroed A columns.
__global__ __launch_bounds__(128) void ss2d_dtproj_wmma(
    const float* __restrict__ xdbl, const float* __restrict__ Wdt,
    const float* __restrict__ dtb, float* __restrict__ delta)
{
    __shared__ float W2[16][9];
    const int tid  = threadIdx.x;
    const int lane = tid & 31;
    const int wave = tid >> 5;
    const int lt   = blockIdx.x * 4 + wave;
    const int d0   = blockIdx.y * 16;           // 6 d-tiles
    const int bk   = blockIdx.z;
    const int k    = bk & 3;
    const int l0   = lt * 16;
    const int m    = lane & 15;
    const int koff = (lane >> 4) * 2;

    for (int i = tid; i < 16 * 8; i += 128) {
        const int row = i >> 3, r = i & 7;
        W2[row][r] = (r < RRANK)
                   ? Wdt[((size_t)k * DDIM + d0 + row) * RRANK + r] : 0.f;
    }
    __syncthreads();

    const float* xrow = xdbl + (size_t)bk * CPAD * LDIM;
    v8f acc = {};
    #pragma unroll
    for (int kk = 0; kk < 2; ++kk) {            // R=6 padded to 8
        const int r = kk * 4 + koff;
        v2f av, bv;
        av.x = W2[m][r];
        av.y = W2[m][r + 1];
        bv.x = xrow[(size_t)r       * LDIM + l0 + m];
        bv.y = xrow[(size_t)(r + 1) * LDIM + l0 + m];
        acc = __builtin_amdgcn_wmma_f32_16x16x4_f32(false, av, false, bv,
                                                    (short)0, acc, false, false);
    }
    const int mhi = (lane >> 4) << 3;
    float* drow = delta + (size_t)bk * DDIM * LDIM;
    #pragma unroll
    for (int r2 = 0; r2 < 8; ++r2) {
        const int d = d0 + r2 + mhi;
        drow[(size_t)d * LDIM + l0 + m] = softplus_f(acc[r2] + dtb[k * DDIM + d]);
    }
}

// ---------------- K3: phase 1 — per-chunk local scan (h0 = 0) -> q, sum(delta)
__global__ __launch_bounds__(96) void ss2d_scan_phase1(
    const float* __restrict__ delta, const float* __restrict__ x,
    const float* __restrict__ xdbl, const float* __restrict__ A_logs,
    float* __restrict__ qbuf, float* __restrict__ dsumbuf)
{
    __shared__ float Bs[NST][TCHUNK];
    const int c   = blockIdx.x;
    const int bk  = blockIdx.y;
    const int b   = bk >> 2, k = bk & 3;
    const int t0  = c * TCHUNK;
    const int tid = threadIdx.x;                // = d, 0..95

    const float* brow = xdbl + ((size_t)bk * CPAD + RRANK) * LDIM + t0;
    for (int i = tid; i < NST * TCHUNK; i += 96)
        Bs[i >> 7][i & 127] = brow[(size_t)(i >> 7) * LDIM + (i & 127)];
    __syncthreads();

    const int d = tid;
    float a[NST], h[NST];
    #pragma unroll
    for (int n = 0; n < NST; ++n) {
        a[n] = -__expf(A_logs[((size_t)k * DDIM + d) * NST + n]);
        h[n] = 0.f;
    }
    const float* drow = delta + ((size_t)bk * DDIM + d) * LDIM;
    const float* xr   = x + ((size_t)b * DDIM + d) * LDIM;
    float dsum = 0.f;
    for (int tt = 0; tt < TCHUNK; ++tt) {
        const int   t  = t0 + tt;
        const float dl = drow[t];
        const float u  = xr[xs_off(k, t)];
        dsum += dl;
        const float du = dl * u;
        #pragma unroll
        for (int n = 0; n < NST; ++n)
            h[n] = h[n] * __expf(dl * a[n]) + du * Bs[n][tt];
    }
    const size_t base = ((size_t)bk * NCHUNK + c) * DDIM + d;
    dsumbuf[base] = dsum;
    #pragma unroll
    for (int n = 0; n < NST; ++n) qbuf[base * NST + n] = h[n];
}

// ---------------- K4: phase 2 — inter-chunk prefix (P_c = exp(A * sum(delta)))
__global__ __launch_bounds__(96) void ss2d_scan_phase2(
    const float* __restrict__ A_logs, const float* __restrict__ qbuf,
    const float* __restrict__ dsumbuf, float* __restrict__ hstart)
{
    const int bk = blockIdx.x;
    const int k  = bk & 3;
    const int d  = threadIdx.x;
    float a[NST], hs[NST];
    #pragma unroll
    for (int n = 0; n < NST; ++n) {
        a[n]  = -__expf(A_logs[((size_t)k * DDIM + d) * NST + n]);
        hs[n] = 0.f;
    }
    for (int c = 0; c < NCHUNK; ++c) {
        const size_t base = ((size_t)bk * NCHUNK + c) * DDIM + d;
        const float  ds   = dsumbuf[base];
        #pragma unroll
        for (int n = 0; n < NST; ++n) {
            hstart[base * NST + n] = hs[n];
            hs[n] = hs[n] * __expf(a[n] * ds) + qbuf[base * NST + n];
        }
    }
}

// ---------------- K5: phase 3 — replay from hstart, emit y, merged via atomics
__global__ __launch_bounds__(96) void ss2d_scan_phase3(
    const float* __restrict__ delta, const float* __restrict__ x,
    const float* __restrict__ xdbl, const float* __restrict__ A_logs,
    const float* __restrict__ Dsk, const float* __restrict__ hstart,
    float* __restrict__ ybuf)
{
    __shared__ float Bs[NST][TCHUNK];
    __shared__ float Cs[NST][TCHUNK];
    const int c   = blockIdx.x;
    const int bk  = blockIdx.y;
    const int b   = bk >> 2, k = bk & 3;
    const int t0  = c * TCHUNK;
    const int tid = threadIdx.x;

    const float* brow = xdbl + ((size_t)bk * CPAD + RRANK)       * LDIM + t0;
    const float* crow = xdbl + ((size_t)bk * CPAD + RRANK + NST) * LDIM + t0;
    for (int i = tid; i < NST * TCHUNK; i += 96) {
        Bs[i >> 7][i & 127] = brow[(size_t)(i >> 7) * LDIM + (i & 127)];
        Cs[i >> 7][i & 127] = crow[(size_t)(i >> 7) * LDIM + (i & 127)];
    }
    __syncthreads();

    const int d = tid;
    const size_t base = ((size_t)bk * NCHUNK + c) * DDIM + d;
    float a[NST], h[NST];
    #pragma unroll
    for (int n = 0; n < NST; ++n) {
        a[n] = -__expf(A_logs[((size_t)k * DDIM + d) * NST + n]);
        h[n] = hstart[base * NST + n];
    }
    const float  dskip = Dsk[k * DDIM + d];
    const float* drow  = delta + ((size_t)bk * DDIM + d) * LDIM;
    const float* xr    = x + ((size_t)b * DDIM + d) * LDIM;
    float* yrow = ybuf + ((size_t)b * DDIM + d) * LDIM;
    for (int tt = 0; tt < TCHUNK; ++tt) {
        const int   t  = t0 + tt;
        const float dl = drow[t];
        const int   go = xs_off(k, t);          // gather AND merge-scatter index
        const float u  = xr[go];
        const float du = dl * u;
        float y = 0.f;
        #pragma unroll
        for (int n = 0; n < NST; ++n) {
            h[n] = h[n] * __expf(dl * a[n]) + du * Bs[n][tt];
            y += h[n] * Cs[n][tt];
        }
        atomicAdd(&yrow[go], y + dskip * u);    // CrossMerge
    }
}

__global__ void ss2d_zero(float* __restrict__ p, int n) {
    int i = blockIdx.x * blockDim.x + threadIdx.x;
    if (i < n) p[i] = 0.f;
}

// ---------------- K6: LayerNorm over D=96 channels, write (B,H,W,D) ----------
__global__ __launch_bounds__(256) void ss2d_ln(
    const float* __restrict__ ybuf, const float* __restrict__ lnw,
    const float* __restrict__ lnb, float* __restrict__ out)
{
    const int idx = blockIdx.x * 256 + threadIdx.x;     // b*L + l, 0..16383
    const int b = idx >> 12;
    const int l = idx & (LDIM - 1);
    const float* yb = ybuf + (size_t)b * DDIM * LDIM + l;
    float mu = 0.f;
    for (int dd = 0; dd < DDIM; ++dd) mu += yb[(size_t)dd * LDIM];
    mu *= (1.f / DDIM);
    float var = 0.f;
    for (int dd = 0; dd < DDIM; ++dd) {
        const float tv = yb[(size_t)dd * LDIM] - mu;
        var += tv * tv;
    }
    var *= (1.f / DDIM);
    const float inv = rsqrtf(var + 1e-5f);
    float* o = out + (size_t)idx * DDIM;                // (b*H+h)*W+w == b*L+l
    for (int dd = 0; dd < DDIM; ++dd)
        o[dd] = (yb[(size_t)dd * LDIM] - mu) * inv * lnw[dd] + lnb[dd];
}

extern "C" void kernel_launch(void* const* d_in, const int* in_sizes, int n_in,
                              void* d_out, int out_size, void* d_ws, size_t ws_size,
                              hipStream_t stream)
{
    const float* x    = (const float*)d_in[0];
    const float* Wp   = (const float*)d_in[1];
    const float* Wdt  = (const float*)d_in[2];
    const float* dtb  = (const float*)d_in[3];
    const float* Alog = (const float*)d_in[4];
    const float* Dsk  = (const float*)d_in[5];
    const float* lnw  = (const float*)d_in[6];
    const float* lnb  = (const float*)d_in[7];
    float* out = (float*)d_out;
    (void)in_sizes; (void)n_in; (void)out_size; (void)ws_size;

    char* ws = (char*)d_ws;
    size_t off = 0;
    auto carve = [&](size_t bytes) {
        char* p = ws + off;
        off += (bytes + 255) & ~(size_t)255;
        return p;
    };
    float* xdbl  = (float*)carve((size_t)16 * CPAD * LDIM * 4);          // 12.6 MB
    float* delta = (float*)carve((size_t)16 * DDIM * LDIM * 4);          // 25.2 MB
    float* qbuf  = (float*)carve((size_t)16 * NCHUNK * DDIM * NST * 4);  // 3.1 MB
    float* dsum  = (float*)carve((size_t)16 * NCHUNK * DDIM * 4);        // 0.2 MB
    float* hst   = (float*)carve((size_t)16 * NCHUNK * DDIM * NST * 4);  // 3.1 MB
    float* ybuf  = (float*)carve((size_t)BDIM * DDIM * LDIM * 4);        // 6.3 MB

    ss2d_xproj_wmma <<<dim3(64, 3, 16), 128, 0, stream>>>(x, Wp, xdbl);
    ss2d_dtproj_wmma<<<dim3(64, 6, 16), 128, 0, stream>>>(xdbl, Wdt, dtb, delta);
    ss2d_scan_phase1<<<dim3(NCHUNK, 16), 96, 0, stream>>>(delta, x, xdbl, Alog, qbuf, dsum);
    ss2d_scan_phase2<<<dim3(16),         96, 0, stream>>>(Alog, qbuf, dsum, hst);
    const int yn = BDIM * DDIM * LDIM;
    ss2d_zero       <<<(yn + 255) / 256, 256, 0, stream>>>(ybuf, yn);
    ss2d_scan_phase3<<<dim3(NCHUNK, 16), 96, 0, stream>>>(delta, x, xdbl, Alog, Dsk, hst, ybuf);
    ss2d_ln         <<<dim3(64),        256, 0, stream>>>(ybuf, lnw, lnb, out);
}